// AttentionMambaBlock_25477746000221
// MI455X (gfx1250) — compile-verified
//
#include <hip/hip_runtime.h>
#include <math.h>

#define DIMC 256
#define NHH  8
#define DEPTH 6
#define BB   2
#define HH   48
#define WW   48
#define DI   512
#define NSS  16
#define DCV  4
#define DRR  16
#define LLEN (HH*WW)     /* 2304 */
#define MTOT (BB*LLEN)   /* 4608 */
#define DBLW (DRR+2*NSS) /* 48 */
#define CH   64          /* scan chunk length */
#define NG   (LLEN/CH)   /* 36 chunks */

typedef __attribute__((ext_vector_type(2))) float v2f;
typedef __attribute__((ext_vector_type(8))) float v8f;

__device__ __forceinline__ float siluf(float x) { return x / (1.0f + expf(-x)); }
__device__ __forceinline__ float softplusf(float x) { return x > 20.0f ? x : log1pf(expf(x)); }

// ---------------------------------------------------------------------------
// xt[m][c] = x[b][c][i][j],  m = b*LLEN + i*WW + j
// ---------------------------------------------------------------------------
__global__ void k_transpose(const float* __restrict__ x, float* __restrict__ xt) {
    int tid = blockIdx.x * blockDim.x + threadIdx.x;
    if (tid >= MTOT * DIMC) return;
    int c = tid % DIMC;
    int m = tid / DIMC;
    int b = m / LLEN, l = m % LLEN;
    xt[tid] = x[((size_t)b * DIMC + c) * LLEN + l];
}

// ---------------------------------------------------------------------------
// Tiled fp32 WMMA GEMM: out[m][n] = sum_k A[m][k]*W[n][k] + bias[n]
// One wave computes a 32(M) x 64(N) tile: 2 A-frags x 4 B-frags -> 8 WMMAs
// per K-step of 4 (6 b64 loads per 8 WMMAs). Block = 4 waves = 128(M) x 64(N).
// Requires M % 128 == 0, N % 64 == 0, Kd % 4 == 0.
// ---------------------------------------------------------------------------
__global__ void k_gemm_t(const float* __restrict__ A, const float* __restrict__ W,
                         const float* __restrict__ bias, float* __restrict__ out,
                         int N, int Kd) {
    int lane = threadIdx.x;
    int half = lane >> 4;      // 0: K {0,1}   1: K {2,3}
    int l    = lane & 15;
    int kb   = half * 2;
    int n0 = blockIdx.x * 64;
    int m0 = (blockIdx.y * 4 + threadIdx.y) * 32;
    const float* a0 = A + (size_t)(m0 + l) * Kd + kb;
    const float* a1 = A + (size_t)(m0 + 16 + l) * Kd + kb;
    const float* w0 = W + (size_t)(n0 + l) * Kd + kb;
    const float* w1 = w0 + (size_t)16 * Kd;
    const float* w2 = w0 + (size_t)32 * Kd;
    const float* w3 = w0 + (size_t)48 * Kd;
    v8f acc[2][4];
#pragma unroll
    for (int mt = 0; mt < 2; ++mt)
#pragma unroll
        for (int nt = 0; nt < 4; ++nt)
            acc[mt][nt] = (v8f){0.f, 0.f, 0.f, 0.f, 0.f, 0.f, 0.f, 0.f};

    for (int k = 0; k < Kd; k += 4) {
        v2f A0 = { a0[k], a0[k + 1] };
        v2f A1 = { a1[k], a1[k + 1] };
        v2f B0 = { w0[k], w0[k + 1] };
        v2f B1 = { w1[k], w1[k + 1] };
        v2f B2 = { w2[k], w2[k + 1] };
        v2f B3 = { w3[k], w3[k + 1] };
        acc[0][0] = __builtin_amdgcn_wmma_f32_16x16x4_f32(false, A0, false, B0, (short)0, acc[0][0], false, false);
        acc[0][1] = __builtin_amdgcn_wmma_f32_16x16x4_f32(false, A0, false, B1, (short)0, acc[0][1], false, false);
        acc[0][2] = __builtin_amdgcn_wmma_f32_16x16x4_f32(false, A0, false, B2, (short)0, acc[0][2], false, false);
        acc[0][3] = __builtin_amdgcn_wmma_f32_16x16x4_f32(false, A0, false, B3, (short)0, acc[0][3], false, false);
        acc[1][0] = __builtin_amdgcn_wmma_f32_16x16x4_f32(false, A1, false, B0, (short)0, acc[1][0], false, false);
        acc[1][1] = __builtin_amdgcn_wmma_f32_16x16x4_f32(false, A1, false, B1, (short)0, acc[1][1], false, false);
        acc[1][2] = __builtin_amdgcn_wmma_f32_16x16x4_f32(false, A1, false, B2, (short)0, acc[1][2], false, false);
        acc[1][3] = __builtin_amdgcn_wmma_f32_16x16x4_f32(false, A1, false, B3, (short)0, acc[1][3], false, false);
    }
#pragma unroll
    for (int mt = 0; mt < 2; ++mt) {
        int mr = m0 + mt * 16 + half * 8;
#pragma unroll
        for (int nt = 0; nt < 4; ++nt) {
            int ncol = n0 + nt * 16 + l;
            float bv = bias ? bias[ncol] : 0.0f;
#pragma unroll
            for (int r = 0; r < 8; ++r)
                out[(size_t)(mr + r) * N + ncol] = acc[mt][nt][r] + bv;
        }
    }
}

// ---------------------------------------------------------------------------
// Small WMMA GEMM (16x16 tile per wave) for N not a multiple of 64 (Wx, N=48).
// ---------------------------------------------------------------------------
__global__ void k_gemm(const float* __restrict__ A, const float* __restrict__ W,
                       const float* __restrict__ bias, float* __restrict__ out,
                       int N, int Kd) {
    int lane = threadIdx.x;
    int half = lane >> 4;
    int l    = lane & 15;
    int n0 = blockIdx.x * 16;
    int m0 = (blockIdx.y * blockDim.y + threadIdx.y) * 16;
    const float* arow = A + (size_t)(m0 + l) * Kd;
    const float* wrow = W + (size_t)(n0 + l) * Kd;
    int kb = half * 2;
    v8f acc = {0.f, 0.f, 0.f, 0.f, 0.f, 0.f, 0.f, 0.f};
#pragma unroll 4
    for (int k = 0; k < Kd; k += 4) {
        v2f a = { arow[k + kb], arow[k + kb + 1] };
        v2f b = { wrow[k + kb], wrow[k + kb + 1] };
        acc = __builtin_amdgcn_wmma_f32_16x16x4_f32(false, a, false, b,
                                                    (short)0, acc, false, false);
    }
    int ncol = n0 + l;
    float bv = bias ? bias[ncol] : 0.0f;
    int mrow = m0 + half * 8;
#pragma unroll
    for (int r = 0; r < 8; ++r)
        out[(size_t)(mrow + r) * N + ncol] = acc[r] + bv;
}

// ---------------------------------------------------------------------------
// Neighborhood attention (K=3) with relative position bias.
// One thread per (b, h, i, j); hd = 32.
// ---------------------------------------------------------------------------
__global__ void k_attn(const float* __restrict__ qkv, const float* __restrict__ rpb,
                       float* __restrict__ att) {
    int tid = blockIdx.x * blockDim.x + threadIdx.x;
    if (tid >= BB * NHH * HH * WW) return;
    int j = tid % WW;
    int i = (tid / WW) % HH;
    int h = (tid / (WW * HH)) % NHH;
    int b = tid / (WW * HH * NHH);
    int si = i - 1; si = si < 0 ? 0 : (si > HH - 3 ? HH - 3 : si);
    int sj = j - 1; sj = sj < 0 ? 0 : (sj > WW - 3 ? WW - 3 : sj);
    int m = (b * HH + i) * WW + j;
    const float scale = 0.17677669529663687f;  // 32^-0.5
    float q[32];
    const float* qp = qkv + (size_t)m * (3 * DIMC) + h * 32;
#pragma unroll
    for (int e = 0; e < 32; ++e) q[e] = qp[e] * scale;

    float lg[9];
    float mx = -1e30f;
    for (int x = 0; x < 3; ++x)
        for (int y = 0; y < 3; ++y) {
            int mn = (b * HH + si + x) * WW + (sj + y);
            const float* kp = qkv + (size_t)mn * (3 * DIMC) + DIMC + h * 32;
            float s = rpb[(h * 5 + (si + x - i + 2)) * 5 + (sj + y - j + 2)];
#pragma unroll
            for (int e = 0; e < 32; ++e) s += q[e] * kp[e];
            lg[x * 3 + y] = s;
            mx = s > mx ? s : mx;
        }
    float den = 0.f;
#pragma unroll
    for (int t = 0; t < 9; ++t) { lg[t] = expf(lg[t] - mx); den += lg[t]; }
    float inv = 1.0f / den;
    float o[32];
#pragma unroll
    for (int e = 0; e < 32; ++e) o[e] = 0.f;
    for (int x = 0; x < 3; ++x)
        for (int y = 0; y < 3; ++y) {
            int mn = (b * HH + si + x) * WW + (sj + y);
            const float* vp = qkv + (size_t)mn * (3 * DIMC) + 2 * DIMC + h * 32;
            float a = lg[x * 3 + y] * inv;
#pragma unroll
            for (int e = 0; e < 32; ++e) o[e] += a * vp[e];
        }
    float* op = att + (size_t)m * DIMC + h * 32;
#pragma unroll
    for (int e = 0; e < 32; ++e) op[e] = o[e];
}

// ---------------------------------------------------------------------------
// Causal depth-4 conv along L (per batch) + SiLU.  Reads xc = xz[:, :DI].
// ---------------------------------------------------------------------------
__global__ void k_conv(const float* __restrict__ xz, const float* __restrict__ cw,
                       const float* __restrict__ cb, float* __restrict__ xs) {
    int tid = blockIdx.x * blockDim.x + threadIdx.x;
    if (tid >= MTOT * DI) return;
    int c = tid % DI;
    int m = tid / DI;
    int b = m / LLEN, l = m % LLEN;
    float s = cb[c];
#pragma unroll
    for (int t = 0; t < DCV; ++t) {
        int l2 = l - (DCV - 1) + t;
        if (l2 >= 0)
            s += xz[((size_t)(b * LLEN + l2)) * (2 * DI) + c] * cw[c * DCV + t];
    }
    xs[tid] = siluf(s);
}

// ---------------------------------------------------------------------------
// Chunked selective scan, pass 1: per (b, d, chunk) compute the chunk's
// transition:  P[n] = prod dA,  hloc[n] = end state starting from h=0.
// ---------------------------------------------------------------------------
__global__ void k_scan_part(const float* __restrict__ dbl, const float* __restrict__ xs,
                            const float* __restrict__ Wdt, const float* __restrict__ bdt,
                            const float* __restrict__ A_log,
                            float* __restrict__ Pbuf, float* __restrict__ Hbuf) {
    int tid = blockIdx.x * blockDim.x + threadIdx.x;
    if (tid >= BB * DI * NG) return;
    int g = tid % NG;
    int r2 = tid / NG;
    int d = r2 % DI, b = r2 / DI;
    float Av[NSS], Wv[NSS], P[NSS], h[NSS];
#pragma unroll
    for (int n = 0; n < NSS; ++n) {
        Av[n] = -expf(A_log[d * NSS + n]);
        Wv[n] = Wdt[d * DRR + n];
        P[n] = 1.f; h[n] = 0.f;
    }
    float bd = bdt[d];
    const float* db = dbl + ((size_t)b * LLEN + g * CH) * DBLW;
    const float* xb = xs  + ((size_t)b * LLEN + g * CH) * DI + d;
    for (int t = 0; t < CH; ++t) {
        const float* row = db + (size_t)t * DBLW;
        float s = bd;
#pragma unroll
        for (int r = 0; r < DRR; ++r) s += row[r] * Wv[r];
        float delta = softplusf(s);
        float dxu = delta * xb[(size_t)t * DI];
#pragma unroll
        for (int n = 0; n < NSS; ++n) {
            float dA = expf(delta * Av[n]);
            P[n] *= dA;
            h[n] = dA * h[n] + dxu * row[DRR + n];
        }
    }
    float* pp = Pbuf + (size_t)tid * NSS;
    float* hp = Hbuf + (size_t)tid * NSS;
#pragma unroll
    for (int n = 0; n < NSS; ++n) { pp[n] = P[n]; hp[n] = h[n]; }
}

// ---------------------------------------------------------------------------
// Pass 2: per (b, d) sequential fixup over 36 chunks.  Overwrites Pbuf with
// the true chunk-start state Hs[g] (exact recurrence across chunk summaries).
// ---------------------------------------------------------------------------
__global__ void k_scan_fix(float* __restrict__ Pbuf, const float* __restrict__ Hbuf) {
    int tid = blockIdx.x * blockDim.x + threadIdx.x;
    if (tid >= BB * DI) return;
    float h[NSS];
#pragma unroll
    for (int n = 0; n < NSS; ++n) h[n] = 0.f;
    for (int g = 0; g < NG; ++g) {
        size_t o = ((size_t)tid * NG + g) * NSS;
#pragma unroll
        for (int n = 0; n < NSS; ++n) {
            float P  = Pbuf[o + n];
            float hl = Hbuf[o + n];
            float hs = h[n];
            Pbuf[o + n] = hs;           // chunk-start state
            h[n] = P * hs + hl;         // propagate to next chunk
        }
    }
}

// ---------------------------------------------------------------------------
// Pass 3: per (b, d, chunk) rerun chunk from its true start state, emit y.
// ---------------------------------------------------------------------------
__global__ void k_scan_chunk(const float* __restrict__ dbl, const float* __restrict__ xs,
                             const float* __restrict__ Wdt, const float* __restrict__ bdt,
                             const float* __restrict__ A_log,
                             const float* __restrict__ Hs, float* __restrict__ ys) {
    int tid = blockIdx.x * blockDim.x + threadIdx.x;
    if (tid >= BB * DI * NG) return;
    int g = tid % NG;
    int r2 = tid / NG;
    int d = r2 % DI, b = r2 / DI;
    float Av[NSS], Wv[NSS], h[NSS];
    const float* hsp = Hs + (size_t)tid * NSS;
#pragma unroll
    for (int n = 0; n < NSS; ++n) {
        Av[n] = -expf(A_log[d * NSS + n]);
        Wv[n] = Wdt[d * DRR + n];
        h[n]  = hsp[n];
    }
    float bd = bdt[d];
    const float* db = dbl + ((size_t)b * LLEN + g * CH) * DBLW;
    const float* xb = xs  + ((size_t)b * LLEN + g * CH) * DI + d;
    float*       yb = ys  + ((size_t)b * LLEN + g * CH) * DI + d;
    for (int t = 0; t < CH; ++t) {
        const float* row = db + (size_t)t * DBLW;
        float s = bd;
#pragma unroll
        for (int r = 0; r < DRR; ++r) s += row[r] * Wv[r];
        float delta = softplusf(s);
        float dxu = delta * xb[(size_t)t * DI];
        float y = 0.f;
#pragma unroll
        for (int n = 0; n < NSS; ++n) {
            float dA = expf(delta * Av[n]);
            h[n] = dA * h[n] + dxu * row[DRR + n];
            y += h[n] * row[DRR + NSS + n];
        }
        yb[(size_t)t * DI] = y;
    }
}

// ---------------------------------------------------------------------------
// Gate: ys = (ys + xs*Dp) * silu(z),  z = xz[:, DI + d]
// ---------------------------------------------------------------------------
__global__ void k_gate(const float* __restrict__ xz, const float* __restrict__ xs,
                       const float* __restrict__ Dp, float* __restrict__ ys) {
    int tid = blockIdx.x * blockDim.x + threadIdx.x;
    if (tid >= MTOT * DI) return;
    int d = tid % DI;
    int m = tid / DI;
    float z = xz[(size_t)m * (2 * DI) + DI + d];
    ys[tid] = (ys[tid] + xs[tid] * Dp[d]) * siluf(z);
}

// ---------------------------------------------------------------------------
// RMSNorm, in place on u (one block per row of 256).
// ---------------------------------------------------------------------------
__global__ void k_rms(float* __restrict__ u, const float* __restrict__ wrms) {
    __shared__ float red[DIMC];
    int m = blockIdx.x;
    int c = threadIdx.x;
    float v = u[(size_t)m * DIMC + c];
    red[c] = v * v;
    __syncthreads();
    for (int s = DIMC / 2; s > 0; s >>= 1) {
        if (c < s) red[c] += red[c + s];
        __syncthreads();
    }
    float inv = rsqrtf(red[0] / (float)DIMC + 1e-6f);
    u[(size_t)m * DIMC + c] = v * inv * wrms[c];
}

// ---------------------------------------------------------------------------
// Final tiled GEMM over concat([xa, u]) with transposed (b, o, i, j) store.
// out[b][o][l] = sum_c cat[m][c] * Wc[o][c] + bc[o]
// ---------------------------------------------------------------------------
__global__ void k_gemm_cat_t(const float* __restrict__ A1, const float* __restrict__ A2,
                             const float* __restrict__ W, const float* __restrict__ bias,
                             float* __restrict__ out) {
    int lane = threadIdx.x;
    int half = lane >> 4, l = lane & 15;
    int kb = half * 2;
    int n0 = blockIdx.x * 64;
    int m0 = (blockIdx.y * 4 + threadIdx.y) * 32;
    const float* a10 = A1 + (size_t)(m0 + l) * DIMC + kb;
    const float* a11 = A1 + (size_t)(m0 + 16 + l) * DIMC + kb;
    const float* a20 = A2 + (size_t)(m0 + l) * DIMC + kb;
    const float* a21 = A2 + (size_t)(m0 + 16 + l) * DIMC + kb;
    const float* w0 = W + (size_t)(n0 + l) * (2 * DIMC) + kb;
    const float* w1 = w0 + (size_t)16 * (2 * DIMC);
    const float* w2 = w0 + (size_t)32 * (2 * DIMC);
    const float* w3 = w0 + (size_t)48 * (2 * DIMC);
    v8f acc[2][4];
#pragma unroll
    for (int mt = 0; mt < 2; ++mt)
#pragma unroll
        for (int nt = 0; nt < 4; ++nt)
            acc[mt][nt] = (v8f){0.f, 0.f, 0.f, 0.f, 0.f, 0.f, 0.f, 0.f};

    for (int k = 0; k < DIMC; k += 4) {
        v2f A0 = { a10[k], a10[k + 1] };
        v2f A1v = { a11[k], a11[k + 1] };
        v2f B0 = { w0[k], w0[k + 1] };
        v2f B1 = { w1[k], w1[k + 1] };
        v2f B2 = { w2[k], w2[k + 1] };
        v2f B3 = { w3[k], w3[k + 1] };
        acc[0][0] = __builtin_amdgcn_wmma_f32_16x16x4_f32(false, A0, false, B0, (short)0, acc[0][0], false, false);
        acc[0][1] = __builtin_amdgcn_wmma_f32_16x16x4_f32(false, A0, false, B1, (short)0, acc[0][1], false, false);
        acc[0][2] = __builtin_amdgcn_wmma_f32_16x16x4_f32(false, A0, false, B2, (short)0, acc[0][2], false, false);
        acc[0][3] = __builtin_amdgcn_wmma_f32_16x16x4_f32(false, A0, false, B3, (short)0, acc[0][3], false, false);
        acc[1][0] = __builtin_amdgcn_wmma_f32_16x16x4_f32(false, A1v, false, B0, (short)0, acc[1][0], false, false);
        acc[1][1] = __builtin_amdgcn_wmma_f32_16x16x4_f32(false, A1v, false, B1, (short)0, acc[1][1], false, false);
        acc[1][2] = __builtin_amdgcn_wmma_f32_16x16x4_f32(false, A1v, false, B2, (short)0, acc[1][2], false, false);
        acc[1][3] = __builtin_amdgcn_wmma_f32_16x16x4_f32(false, A1v, false, B3, (short)0, acc[1][3], false, false);
    }
    for (int k = 0; k < DIMC; k += 4) {
        v2f A0 = { a20[k], a20[k + 1] };
        v2f A1v = { a21[k], a21[k + 1] };
        v2f B0 = { w0[DIMC + k], w0[DIMC + k + 1] };
        v2f B1 = { w1[DIMC + k], w1[DIMC + k + 1] };
        v2f B2 = { w2[DIMC + k], w2[DIMC + k + 1] };
        v2f B3 = { w3[DIMC + k], w3[DIMC + k + 1] };
        acc[0][0] = __builtin_amdgcn_wmma_f32_16x16x4_f32(false, A0, false, B0, (short)0, acc[0][0], false, false);
        acc[0][1] = __builtin_amdgcn_wmma_f32_16x16x4_f32(false, A0, false, B1, (short)0, acc[0][1], false, false);
        acc[0][2] = __builtin_amdgcn_wmma_f32_16x16x4_f32(false, A0, false, B2, (short)0, acc[0][2], false, false);
        acc[0][3] = __builtin_amdgcn_wmma_f32_16x16x4_f32(false, A0, false, B3, (short)0, acc[0][3], false, false);
        acc[1][0] = __builtin_amdgcn_wmma_f32_16x16x4_f32(false, A1v, false, B0, (short)0, acc[1][0], false, false);
        acc[1][1] = __builtin_amdgcn_wmma_f32_16x16x4_f32(false, A1v, false, B1, (short)0, acc[1][1], false, false);
        acc[1][2] = __builtin_amdgcn_wmma_f32_16x16x4_f32(false, A1v, false, B2, (short)0, acc[1][2], false, false);
        acc[1][3] = __builtin_amdgcn_wmma_f32_16x16x4_f32(false, A1v, false, B3, (short)0, acc[1][3], false, false);
    }
#pragma unroll
    for (int mt = 0; mt < 2; ++mt) {
        int mr = m0 + mt * 16 + half * 8;
#pragma unroll
        for (int nt = 0; nt < 4; ++nt) {
            int ncol = n0 + nt * 16 + l;
            float bv = bias[ncol];
#pragma unroll
            for (int r = 0; r < 8; ++r) {
                int m = mr + r;
                int b_ = m / LLEN, li = m % LLEN;
                out[((size_t)b_ * (2 * DIMC) + ncol) * LLEN + li] = acc[mt][nt][r] + bv;
            }
        }
    }
}

// ---------------------------------------------------------------------------
extern "C" void kernel_launch(void* const* d_in, const int* in_sizes, int n_in,
                              void* d_out, int out_size, void* d_ws, size_t ws_size,
                              hipStream_t stream) {
    (void)in_sizes; (void)n_in; (void)out_size; (void)ws_size;
    const float* x     = (const float*)d_in[0];
    const float* Wqkv  = (const float*)d_in[1];
    const float* bqkv  = (const float*)d_in[2];
    const float* rpb   = (const float*)d_in[3];
    const float* Wpr   = (const float*)d_in[4];
    const float* bpr   = (const float*)d_in[5];
    const float* Win   = (const float*)d_in[6];
    const float* convw = (const float*)d_in[7];
    const float* convb = (const float*)d_in[8];
    const float* Wx    = (const float*)d_in[9];
    const float* Wdt   = (const float*)d_in[10];
    const float* bdt   = (const float*)d_in[11];
    const float* A_log = (const float*)d_in[12];
    const float* Dp    = (const float*)d_in[13];
    const float* Wout  = (const float*)d_in[14];
    const float* wrms  = (const float*)d_in[15];
    const float* Wc    = (const float*)d_in[16];
    const float* bc    = (const float*)d_in[17];

    float* ws   = (float*)d_ws;
    float* xt   = ws;                                   // 4608*256
    float* qkv  = xt  + (size_t)MTOT * DIMC;            // 4608*768
    float* att  = qkv + (size_t)MTOT * 3 * DIMC;        // 4608*256
    float* xa   = att + (size_t)MTOT * DIMC;            // 4608*256
    float* u    = xa  + (size_t)MTOT * DIMC;            // 4608*256
    float* xz   = u   + (size_t)MTOT * DIMC;            // 4608*1024
    float* xs   = xz  + (size_t)MTOT * 2 * DI;          // 4608*512
    float* dbl  = xs  + (size_t)MTOT * DI;              // 4608*48
    float* ys   = dbl + (size_t)MTOT * DBLW;            // 4608*512
    float* Pbuf = ys  + (size_t)MTOT * DI;              // 2*512*36*16
    float* Hbuf = Pbuf + (size_t)BB * DI * NG * NSS;    // 2*512*36*16

    dim3 blk(32, 4);                 // 4 waves per block
    const int myT = MTOT / 128;      // 36 blocks along M for tiled GEMM
    const int myS = MTOT / 64;       // 72 blocks along M for small GEMM
    const int scanThreads = BB * DI * NG;  // 36864

    k_transpose<<<(MTOT * DIMC + 255) / 256, 256, 0, stream>>>(x, xt);
    k_gemm_t<<<dim3(3 * DIMC / 64, myT), blk, 0, stream>>>(xt, Wqkv, bqkv, qkv, 3 * DIMC, DIMC);
    k_attn<<<(BB * NHH * HH * WW + 255) / 256, 256, 0, stream>>>(qkv, rpb, att);
    k_gemm_t<<<dim3(DIMC / 64, myT), blk, 0, stream>>>(att, Wpr, bpr, xa, DIMC, DIMC);

    for (int i = 0; i < DEPTH; ++i) {
        const float* uin = (i == 0) ? xa : u;
        k_gemm_t<<<dim3(2 * DI / 64, myT), blk, 0, stream>>>(
            uin, Win + (size_t)i * 2 * DI * DIMC, nullptr, xz, 2 * DI, DIMC);
        k_conv<<<(MTOT * DI + 255) / 256, 256, 0, stream>>>(
            xz, convw + (size_t)i * DI * DCV, convb + (size_t)i * DI, xs);
        k_gemm<<<dim3(DBLW / 16, myS), blk, 0, stream>>>(
            xs, Wx + (size_t)i * DBLW * DI, nullptr, dbl, DBLW, DI);
        k_scan_part<<<(scanThreads + 255) / 256, 256, 0, stream>>>(
            dbl, xs, Wdt + (size_t)i * DI * DRR, bdt + (size_t)i * DI,
            A_log + (size_t)i * DI * NSS, Pbuf, Hbuf);
        k_scan_fix<<<(BB * DI + 255) / 256, 256, 0, stream>>>(Pbuf, Hbuf);
        k_scan_chunk<<<(scanThreads + 255) / 256, 256, 0, stream>>>(
            dbl, xs, Wdt + (size_t)i * DI * DRR, bdt + (size_t)i * DI,
            A_log + (size_t)i * DI * NSS, Pbuf, ys);
        k_gate<<<(MTOT * DI + 255) / 256, 256, 0, stream>>>(
            xz, xs, Dp + (size_t)i * DI, ys);
        k_gemm_t<<<dim3(DIMC / 64, myT), blk, 0, stream>>>(
            ys, Wout + (size_t)i * DIMC * DI, nullptr, u, DIMC, DI);
    }

    k_rms<<<MTOT, DIMC, 0, stream>>>(u, wrms);
    k_gemm_cat_t<<<dim3(2 * DIMC / 64, myT), blk, 0, stream>>>(xa, u, Wc, bc, (float*)d_out);
}